// TransFGDeiT_24850680774682
// MI455X (gfx1250) — compile-verified
//
#include <hip/hip_runtime.h>
#include <hip/hip_bf16.h>
#include <math.h>

typedef __attribute__((ext_vector_type(2))) float v2f;
typedef __attribute__((ext_vector_type(8))) float v8f;

#define B_   64
#define N_   197
#define D_   768
#define H_   12
#define P_   12
#define C_   200
#define NPATCH 196

// ---------------------------------------------------------------------------
// K1: attn_scores (only row-0 slices of attns!) + iterative top-12 per batch.
// attns: (4, B, H, N, N). Need attns[l,b,h,0,1+j], j in [0,196).
// ---------------------------------------------------------------------------
__global__ void topk_kernel(const float* __restrict__ attns, int* __restrict__ idx_out)
{
    __shared__ float s[NPATCH];
    __shared__ float rv[256];
    __shared__ int   ri[256];
    const int b = blockIdx.x;
    const int t = threadIdx.x;

    for (int j = t; j < NPATCH; j += 256) {
        float acc = 0.f;
        for (int l = 0; l < 4; ++l)
            for (int h = 0; h < H_; ++h) {
                size_t base = ((((size_t)l * B_ + b) * H_ + h) * N_) * N_; // row 0
                acc += attns[base + 1 + j];
            }
        s[j] = acc * (1.0f / 48.0f);
    }
    __syncthreads();

    for (int p = 0; p < P_; ++p) {
        float bv = -1e30f; int bi = -1;
        for (int j = t; j < NPATCH; j += 256)
            if (s[j] > bv) { bv = s[j]; bi = j; }
        rv[t] = bv; ri[t] = bi;
        __syncthreads();
        for (int off = 128; off > 0; off >>= 1) {
            if (t < off) {
                bool take = (rv[t + off] > rv[t]) ||
                            (rv[t + off] == rv[t] && ri[t + off] >= 0 &&
                             (ri[t] < 0 || ri[t + off] < ri[t]));
                if (take) { rv[t] = rv[t + off]; ri[t] = ri[t + off]; }
            }
            __syncthreads();
        }
        if (t == 0) { idx_out[b * P_ + p] = ri[0]; s[ri[0]] = -1e30f; }
        __syncthreads();
    }
}

// ---------------------------------------------------------------------------
// Block LayerNorm over a 768-float row (256 threads/row).
// ---------------------------------------------------------------------------
__device__ __forceinline__ void ln_row(const float* __restrict__ x,
                                       const float* __restrict__ g,
                                       const float* __restrict__ bb,
                                       float* __restrict__ out,
                                       float* red, int t)
{
    float ls = 0.f;
    for (int d = t; d < D_; d += 256) ls += x[d];
    red[t] = ls; __syncthreads();
    for (int off = 128; off > 0; off >>= 1) { if (t < off) red[t] += red[t + off]; __syncthreads(); }
    const float mean = red[0] * (1.0f / D_);
    __syncthreads();
    float lv = 0.f;
    for (int d = t; d < D_; d += 256) { float df = x[d] - mean; lv += df * df; }
    red[t] = lv; __syncthreads();
    for (int off = 128; off > 0; off >>= 1) { if (t < off) red[t] += red[t + off]; __syncthreads(); }
    const float inv = 1.0f / sqrtf(red[0] * (1.0f / D_) + 1e-5f);
    __syncthreads();
    for (int d = t; d < D_; d += 256) out[d] = (x[d] - mean) * inv * g[d] + bb[d];
}

// K2: gather selected patches + LN1.  one block per (b,p) row.
__global__ void gather_ln_kernel(const float* __restrict__ hidden,
                                 const int* __restrict__ idx,
                                 const float* __restrict__ g,
                                 const float* __restrict__ bb,
                                 float* __restrict__ out)
{
    __shared__ float red[256];
    const int row = blockIdx.x;            // 0..B*P-1
    const int b = row / P_;
    const int patch = idx[row];
    const float* x = hidden + ((size_t)b * N_ + 1 + patch) * D_;
    ln_row(x, g, bb, out + (size_t)row * D_, red, threadIdx.x);
}

// K6a: plain LN (LN2). one block per row.
__global__ void ln_kernel(const float* __restrict__ in,
                          const float* __restrict__ g,
                          const float* __restrict__ bb,
                          float* __restrict__ out)
{
    __shared__ float red[256];
    const int row = blockIdx.x;
    ln_row(in + (size_t)row * D_, g, bb, out + (size_t)row * D_, red, threadIdx.x);
}

// ---------------------------------------------------------------------------
// WMMA fp32 GEMM:  C[M,N] = A[M,K] * W[N,K]^T + bias[N]   (optional relu)
// One wave per 16x16 tile, K stepped by 4 via V_WMMA_F32_16X16X4_F32.
// Fragment layouts per CDNA5 ISA 7.12.2:
//   A 16x4 : lane<16 -> M=lane, K={k,k+1};  lane>=16 -> M=lane-16, K={k+2,k+3}
//   B 4x16 : lane<16 -> N=lane, K={k,k+1};  lane>=16 -> N=lane-16, K={k+2,k+3}
//   C/D    : vgpr r  -> row tm + r + 8*(lane>=16), col tn + (lane&15)
// Early exit is wave-uniform so EXEC is all-ones around every WMMA.
// ---------------------------------------------------------------------------
__global__ void wmma_gemm_nt(const float* __restrict__ A,
                             const float* __restrict__ W,
                             const float* __restrict__ bias,
                             float* __restrict__ C,
                             int M, int N, int K, int relu)
{
    const int lane  = threadIdx.x & 31;
    const int wave  = (blockIdx.x * blockDim.x + threadIdx.x) >> 5;
    const int tilesN = N >> 4;
    const int tilesM = M >> 4;
    if (wave >= tilesM * tilesN) return;            // wave-uniform exit
    const int tm = (wave / tilesN) << 4;
    const int tn = (wave % tilesN) << 4;

    const int hl  = lane >> 4;                      // 0: lanes 0-15, 1: lanes 16-31
    const int l15 = lane & 15;
    const float* arow = A + (size_t)(tm + l15) * K; // A row for this lane
    const float* wrow = W + (size_t)(tn + l15) * K; // W row n holds B[k][n] along k

    v8f acc = {0.f, 0.f, 0.f, 0.f, 0.f, 0.f, 0.f, 0.f};
    for (int k = 0; k < K; k += 4) {
        const int kk = k + (hl << 1);
        v2f a, b;
        a.x = arow[kk]; a.y = arow[kk + 1];
        b.x = wrow[kk]; b.y = wrow[kk + 1];
        acc = __builtin_amdgcn_wmma_f32_16x16x4_f32(
            /*neg_a=*/false, a, /*neg_b=*/false, b,
            /*c_mod=*/(short)0, acc, /*reuse_a=*/false, /*reuse_b=*/false);
    }
#pragma unroll
    for (int r = 0; r < 8; ++r) {
        const int row = tm + r + (hl << 3);
        const int col = tn + l15;
        float v = acc[r] + (bias ? bias[col] : 0.0f);
        if (relu) v = fmaxf(v, 0.0f);
        C[(size_t)row * N + col] = v;
    }
}

// ---------------------------------------------------------------------------
// K4: per (b,h) attention over the 12 parts. qkv layout (B, P, 3*D).
// ---------------------------------------------------------------------------
__global__ void attn_kernel(const float* __restrict__ qkv, float* __restrict__ obuf)
{
    const int bh = blockIdx.x;
    const int b = bh / H_, h = bh % H_;
    __shared__ float q[P_][64], k[P_][64], v[P_][64];
    __shared__ float sc[P_][P_], aw[P_][P_];
    const int t = threadIdx.x;                      // 256
    const float* base = qkv + (size_t)b * P_ * (3 * D_);

    for (int e = t; e < P_ * 64; e += 256) {
        const int p = e >> 6, d = e & 63;
        q[p][d] = base[(size_t)p * (3 * D_) +          h * 64 + d];
        k[p][d] = base[(size_t)p * (3 * D_) + D_     + h * 64 + d];
        v[p][d] = base[(size_t)p * (3 * D_) + 2 * D_ + h * 64 + d];
    }
    __syncthreads();

    if (t < P_ * P_) {
        const int qi = t / P_, ki = t % P_;
        float a = 0.f;
        for (int d = 0; d < 64; ++d) a += q[qi][d] * k[ki][d];
        sc[qi][ki] = a * 0.125f;                    // 1/sqrt(64)
    }
    __syncthreads();

    if (t < P_) {
        float mx = -1e30f;
        for (int j = 0; j < P_; ++j) mx = fmaxf(mx, sc[t][j]);
        float s = 0.f;
        for (int j = 0; j < P_; ++j) { float e = expf(sc[t][j] - mx); aw[t][j] = e; s += e; }
        const float is = 1.0f / s;
        for (int j = 0; j < P_; ++j) aw[t][j] *= is;
    }
    __syncthreads();

    for (int e = t; e < P_ * 64; e += 256) {
        const int qi = e >> 6, d = e & 63;
        float a = 0.f;
        for (int j = 0; j < P_; ++j) a += aw[qi][j] * v[j][d];
        obuf[((size_t)b * P_ + qi) * D_ + h * 64 + d] = a;
    }
}

// ---------------------------------------------------------------------------
// K6b: softmax pooling per batch -> part_feat (B, D)
// ---------------------------------------------------------------------------
__global__ void pool_kernel(const float* __restrict__ attn_ln,
                            const float* __restrict__ w_pool,
                            const float* __restrict__ b_pool,
                            const float* __restrict__ pool_temp,
                            float* __restrict__ part_feat)
{
    __shared__ float red[256];
    __shared__ float w[P_];
    const int b = blockIdx.x, t = threadIdx.x;
    const float* base = attn_ln + (size_t)b * P_ * D_;

    for (int p = 0; p < P_; ++p) {
        float ls = 0.f;
        for (int d = t; d < D_; d += 256) ls += base[p * D_ + d] * w_pool[d];
        red[t] = ls; __syncthreads();
        for (int off = 128; off > 0; off >>= 1) { if (t < off) red[t] += red[t + off]; __syncthreads(); }
        if (t == 0) w[p] = red[0] + b_pool[0];
        __syncthreads();
    }
    if (t == 0) {
        const float temp = fmaxf(pool_temp[0], 0.3f);
        float mx = -1e30f;
        for (int p = 0; p < P_; ++p) { w[p] /= temp; mx = fmaxf(mx, w[p]); }
        float s = 0.f;
        for (int p = 0; p < P_; ++p) { w[p] = expf(w[p] - mx); s += w[p]; }
        for (int p = 0; p < P_; ++p) w[p] /= s;
    }
    __syncthreads();
    for (int d = t; d < D_; d += 256) {
        float a = 0.f;
        for (int p = 0; p < P_; ++p) a += base[p * D_ + d] * w[p];
        part_feat[(size_t)b * D_ + d] = a;
    }
}

// K8a: inverse L2 norm of fusion = [cls | pf2] per batch
__global__ void fnorm_kernel(const float* __restrict__ hidden,
                             const float* __restrict__ pf2,
                             float* __restrict__ fnorm)
{
    __shared__ float red[256];
    const int b = blockIdx.x, t = threadIdx.x;
    const float* cls = hidden + (size_t)b * N_ * D_;   // row 0
    const float* pf  = pf2 + (size_t)b * D_;
    float ls = 0.f;
    for (int d = t; d < D_; d += 256) ls += cls[d] * cls[d] + pf[d] * pf[d];
    red[t] = ls; __syncthreads();
    for (int off = 128; off > 0; off >>= 1) { if (t < off) red[t] += red[t + off]; __syncthreads(); }
    if (t == 0) fnorm[b] = 1.0f / sqrtf(red[0]);
}

// K8b: inverse L2 norm of each arc_w row (200 x 1536)
__global__ void wnorm_kernel(const float* __restrict__ arc_w, float* __restrict__ wnorm)
{
    __shared__ float red[256];
    const int c = blockIdx.x, t = threadIdx.x;
    const float* w = arc_w + (size_t)c * (2 * D_);
    float ls = 0.f;
    for (int d = t; d < 2 * D_; d += 256) ls += w[d] * w[d];
    red[t] = ls; __syncthreads();
    for (int off = 128; off > 0; off >>= 1) { if (t < off) red[t] += red[t + off]; __syncthreads(); }
    if (t == 0) wnorm[c] = 1.0f / sqrtf(red[0]);
}

// K8c: cosine + ArcFace margin -> logits (B, C)
__global__ void arcface_kernel(const float* __restrict__ hidden,
                               const float* __restrict__ pf2,
                               const float* __restrict__ arc_w,
                               const float* __restrict__ fnorm,
                               const float* __restrict__ wnorm,
                               const int* __restrict__ labels,
                               float* __restrict__ out)
{
    __shared__ float fus[2 * D_];
    const int b = blockIdx.x, t = threadIdx.x;
    const float* cls = hidden + (size_t)b * N_ * D_;
    const float* pf  = pf2 + (size_t)b * D_;
    for (int d = t; d < D_; d += 256) { fus[d] = cls[d]; fus[D_ + d] = pf[d]; }
    __syncthreads();
    if (t < C_) {
        const float* w = arc_w + (size_t)t * (2 * D_);
        float a = 0.f;
        for (int kk = 0; kk < 2 * D_; ++kk) a += fus[kk] * w[kk];
        const float cosine = a * fnorm[b] * wnorm[t];
        const float sine = sqrtf(fmaxf(0.0f, fminf(1.0f, 1.0f - cosine * cosine)));
        const float cos_m = 0.87758256189037276f;   // cos(0.5)
        const float sin_m = 0.47942553860420301f;   // sin(0.5)
        const float th    = -0.87758256189037276f;  // cos(pi-0.5)
        const float mm    = 0.23971276930210151f;   // sin(pi-0.5)*0.5
        float phi = cosine * cos_m - sine * sin_m;
        phi = (cosine > th) ? phi : (cosine - mm);
        const float oh = (labels[b] == t) ? 1.0f : 0.0f;
        out[(size_t)b * C_ + t] = (oh * phi + (1.0f - oh) * cosine) * 30.0f;
    }
}

// ---------------------------------------------------------------------------
extern "C" void kernel_launch(void* const* d_in, const int* in_sizes, int n_in,
                              void* d_out, int out_size, void* d_ws, size_t ws_size,
                              hipStream_t stream) {
    const float* hidden    = (const float*)d_in[0];
    const float* attns     = (const float*)d_in[1];
    const int*   labels    = (const int*)  d_in[2];
    const float* ln1_g     = (const float*)d_in[3];
    const float* ln1_b     = (const float*)d_in[4];
    const float* w_in      = (const float*)d_in[5];
    const float* b_in      = (const float*)d_in[6];
    const float* w_out     = (const float*)d_in[7];
    const float* b_out     = (const float*)d_in[8];
    const float* ln2_g     = (const float*)d_in[9];
    const float* ln2_b     = (const float*)d_in[10];
    const float* w_pool    = (const float*)d_in[11];
    const float* b_pool    = (const float*)d_in[12];
    const float* pool_temp = (const float*)d_in[13];
    const float* w_mlp     = (const float*)d_in[14];
    const float* b_mlp     = (const float*)d_in[15];
    const float* arc_w     = (const float*)d_in[16];
    float* out = (float*)d_out;

    char* ws = (char*)d_ws;
    size_t off = 0;
    auto alloc = [&](size_t bytes) -> void* {
        void* p = ws + off;
        off += (bytes + 255) & ~(size_t)255;
        return p;
    };
    int*   idx       = (int*)  alloc((size_t)B_ * P_ * 4);
    float* parts_ln  = (float*)alloc((size_t)B_ * P_ * D_ * 4);
    float* qkv       = (float*)alloc((size_t)B_ * P_ * 3 * D_ * 4);
    float* obuf      = (float*)alloc((size_t)B_ * P_ * D_ * 4);
    float* attn_out  = (float*)alloc((size_t)B_ * P_ * D_ * 4);
    float* attn_ln   = (float*)alloc((size_t)B_ * P_ * D_ * 4);
    float* part_feat = (float*)alloc((size_t)B_ * D_ * 4);
    float* pf2       = (float*)alloc((size_t)B_ * D_ * 4);
    float* fnorm     = (float*)alloc((size_t)B_ * 4);
    float* wnorm     = (float*)alloc((size_t)C_ * 4);

    // 1. patch scoring + top-12 (reads only 2.4 MB of the 476 MB attns tensor)
    topk_kernel<<<B_, 256, 0, stream>>>(attns, idx);
    // 2. gather + LN1
    gather_ln_kernel<<<B_ * P_, 256, 0, stream>>>(hidden, idx, ln1_g, ln1_b, parts_ln);
    // 3. QKV GEMM: (768 x 2304) = (768 x 768) * w_in^T   [WMMA f32]
    { int tiles = (B_ * P_ / 16) * (3 * D_ / 16);          // 48*144 = 6912
      wmma_gemm_nt<<<tiles / 8, 256, 0, stream>>>(parts_ln, w_in, b_in, qkv,
                                                  B_ * P_, 3 * D_, D_, 0); }
    // 4. 12x12 attention per (b,h)
    attn_kernel<<<B_ * H_, 256, 0, stream>>>(qkv, obuf);
    // 5. out-proj GEMM: (768 x 768) * w_out^T             [WMMA f32]
    { int tiles = (B_ * P_ / 16) * (D_ / 16);              // 48*48 = 2304
      wmma_gemm_nt<<<tiles / 8, 256, 0, stream>>>(obuf, w_out, b_out, attn_out,
                                                  B_ * P_, D_, D_, 0); }
    // 6. LN2 + softmax pooling
    ln_kernel<<<B_ * P_, 256, 0, stream>>>(attn_out, ln2_g, ln2_b, attn_ln);
    pool_kernel<<<B_, 256, 0, stream>>>(attn_ln, w_pool, b_pool, pool_temp, part_feat);
    // 7. MLP GEMM + relu: (64 x 768) * w_mlp^T            [WMMA f32]
    { int tiles = (B_ / 16) * (D_ / 16);                   // 4*48 = 192
      wmma_gemm_nt<<<tiles / 8, 256, 0, stream>>>(part_feat, w_mlp, b_mlp, pf2,
                                                  B_, D_, D_, 1); }
    // 8. ArcFace head
    fnorm_kernel<<<B_, 256, 0, stream>>>(hidden, pf2, fnorm);
    wnorm_kernel<<<C_, 256, 0, stream>>>(arc_w, wnorm);
    arcface_kernel<<<B_, 256, 0, stream>>>(hidden, pf2, arc_w, fnorm, wnorm, labels, out);
}